// GCNRegression_79826262163817
// MI455X (gfx1250) — compile-verified
//
#include <hip/hip_runtime.h>
#include <hip/hip_bf16.h>

typedef float v2f __attribute__((ext_vector_type(2)));
typedef float v8f __attribute__((ext_vector_type(8)));

#define HIDC 16   // hidden width fixed at 16 (= one WMMA tile of columns)

// ---------------- degree / normalization ----------------

__global__ void k_deg_init(float* __restrict__ deg, int n) {
  int i = blockIdx.x * blockDim.x + threadIdx.x;
  if (i < n) deg[i] = 1.0f;  // self-loop weight contributes 1 to every node's degree
}

__global__ void k_deg_accum(const int* __restrict__ col, const float* __restrict__ w,
                            float* __restrict__ deg, int E) {
  int e = blockIdx.x * blockDim.x + threadIdx.x;
  if (e < E) atomicAdd(deg + col[e], w[e]);
}

__global__ void k_deg_rsqrt(float* __restrict__ deg, int n) {
  int i = blockIdx.x * blockDim.x + threadIdx.x;
  if (i < n) {
    float d = deg[i];
    deg[i] = (d > 0.0f) ? rsqrtf(d) : 0.0f;  // becomes dinv in place
  }
}

// ---------------- fp32 WMMA GEMM: C[N x 16] = A[N x K] @ W[K x 16] ----------------
// One wave per 16-row tile. K must be a multiple of 4.
__global__ void k_gemm_wmma(const float* __restrict__ A, const float* __restrict__ W,
                            float* __restrict__ C, int nTiles, int K) {
  int gwave = (int)((blockIdx.x * blockDim.x + threadIdx.x) >> 5);
  int lane  = (int)(threadIdx.x & 31);
  if (gwave >= nTiles) return;          // wave-uniform: surviving waves keep EXEC all-1s

  const int half = lane >> 4;           // 0: lanes 0-15, 1: lanes 16-31
  const int l16  = lane & 15;
  const int m0   = gwave * 16;

  v8f acc = {};
  // A fragment: lane holds A[m0+l16][k0 + 2*half + {0,1}]
  const float* arow = A + (size_t)(m0 + l16) * K + 2 * half;
  // B fragment: lane holds W[k0 + 2*half + {0,1}][l16]
  const float* bcol = W + (size_t)(2 * half) * HIDC + l16;

  for (int k0 = 0; k0 < K; k0 += 4) {
    v2f a, b;
    a.x = arow[0];
    a.y = arow[1];
    b.x = bcol[0];
    b.y = bcol[HIDC];
    acc = __builtin_amdgcn_wmma_f32_16x16x4_f32(
        /*neg_a=*/false, a, /*neg_b=*/false, b,
        /*c_mod=*/(short)0, acc, /*reuse_a=*/false, /*reuse_b=*/false);
    arow += 4;
    bcol += 4 * HIDC;
  }

  // D layout: VGPR v -> row m0 + v + 8*half, col l16
  float* crow = C + (size_t)(m0 + 8 * half) * HIDC + l16;
#pragma unroll
  for (int v = 0; v < 8; ++v) crow[(size_t)v * HIDC] = acc[v];
}

// scalar tail for N % 16 != 0 (not used when N is a multiple of 16)
__global__ void k_gemm_tail(const float* __restrict__ A, const float* __restrict__ W,
                            float* __restrict__ C, int rowStart, int n, int K) {
  int idx = blockIdx.x * blockDim.x + threadIdx.x;
  int r = rowStart + (idx >> 4);
  int c = idx & 15;
  if (r >= n) return;
  float s = 0.0f;
  for (int k = 0; k < K; ++k) s += A[(size_t)r * K + k] * W[(size_t)k * HIDC + c];
  C[(size_t)r * HIDC + c] = s;
}

// ---------------- agg init: self-loop term + bias (full overwrite, no zeroing) ----------------
__global__ void k_self_bias(const float* __restrict__ h, const float* __restrict__ dinv,
                            const float* __restrict__ bias, float* __restrict__ agg, int n) {
  int idx = blockIdx.x * blockDim.x + threadIdx.x;
  if (idx >= n * HIDC) return;
  int i = idx >> 4;
  int c = idx & 15;
  float di = dinv[i];
  agg[idx] = di * di * h[idx] + bias[c];
}

// ---------------- per-edge scatter-add of normalized messages ----------------
__global__ void k_scatter(const int* __restrict__ row, const int* __restrict__ col,
                          const float* __restrict__ w, const float* __restrict__ dinv,
                          const float* __restrict__ h, float* __restrict__ agg, int E) {
  int e = blockIdx.x * blockDim.x + threadIdx.x;
  if (e >= E) return;
  int r = row[e];
  int c = col[e];
  float norm = dinv[r] * w[e] * dinv[c];
  const float4* hr = (const float4*)(h + (size_t)r * HIDC);  // GLOBAL_LOAD_B128 x4
  float* ar = agg + (size_t)c * HIDC;
#pragma unroll
  for (int i = 0; i < 4; ++i) {
    float4 v = hr[i];
    atomicAdd(ar + 4 * i + 0, norm * v.x);
    atomicAdd(ar + 4 * i + 1, norm * v.y);
    atomicAdd(ar + 4 * i + 2, norm * v.z);
    atomicAdd(ar + 4 * i + 3, norm * v.w);
  }
}

__global__ void k_relu(float* __restrict__ a, int total) {
  int idx = blockIdx.x * blockDim.x + threadIdx.x;
  if (idx < total) a[idx] = fmaxf(a[idx], 0.0f);
}

__global__ void k_zero(float* __restrict__ p, int total) {
  int idx = blockIdx.x * blockDim.x + threadIdx.x;
  if (idx < total) p[idx] = 0.0f;
}

// ---------------- global mean pool (accumulate) ----------------
__global__ void k_pool(const float* __restrict__ agg, const int* __restrict__ batch,
                       float* __restrict__ pooled, float* __restrict__ cnt, int n) {
  int idx = blockIdx.x * blockDim.x + threadIdx.x;
  if (idx >= n * HIDC) return;
  int i = idx >> 4;
  int c = idx & 15;
  int g = batch[i];
  atomicAdd(pooled + (size_t)g * HIDC + c, agg[idx]);
  if (c == 0) atomicAdd(cnt + g, 1.0f);
}

// ---------------- head: out[g][o] = (pooled[g]/cnt[g]) . fcW[:,o] + fcb[o] ----------------
__global__ void k_head(const float* __restrict__ pooled, const float* __restrict__ cnt,
                       const float* __restrict__ fcW, const float* __restrict__ fcb,
                       float* __restrict__ out, int G, int OUTC) {
  int idx = blockIdx.x * blockDim.x + threadIdx.x;
  if (idx >= G * OUTC) return;
  int g = idx / OUTC;
  int o = idx % OUTC;
  float inv = 1.0f / fmaxf(cnt[g], 1.0f);
  float s = 0.0f;
#pragma unroll
  for (int c = 0; c < HIDC; ++c) s += pooled[(size_t)g * HIDC + c] * fcW[c * OUTC + o];
  out[idx] = s * inv + fcb[o];
}

extern "C" void kernel_launch(void* const* d_in, const int* in_sizes, int n_in,
                              void* d_out, int out_size, void* d_ws, size_t ws_size,
                              hipStream_t stream) {
  const float* x     = (const float*)d_in[0];
  const int*   eidx  = (const int*)d_in[1];
  const float* ew    = (const float*)d_in[2];
  const int*   batch = (const int*)d_in[3];
  const float* W1    = (const float*)d_in[4];
  const float* b1    = (const float*)d_in[5];
  const float* W2    = (const float*)d_in[6];
  const float* b2    = (const float*)d_in[7];
  const float* fcW   = (const float*)d_in[8];
  const float* fcb   = (const float*)d_in[9];

  const int N    = in_sizes[3];            // 100000 nodes
  const int E    = in_sizes[2];            // 3200000 edges
  const int K1   = in_sizes[0] / N;        // 128 input channels
  const int OUTC = in_sizes[9];            // 1
  const int G    = out_size / OUTC;        // 512 graphs

  const int* row = eidx;       // edge_index[0] = source
  const int* col = eidx + E;   // edge_index[1] = target

  // workspace layout (floats)
  float* dinv   = (float*)d_ws;                       // [N]      deg -> dinv in place
  float* bufA   = dinv + N;                           // [N*16]   h (layer output of GEMM)
  float* bufB   = bufA + (size_t)N * HIDC;            // [N*16]   agg (scatter target)
  float* pooled = bufB + (size_t)N * HIDC;            // [G*16]
  float* cnt    = pooled + (size_t)G * HIDC;          // [G]

  const int T  = 256;                                  // 8 wave32 per block
  const int WPB = T / 32;
  const int nTiles = N / 16;
  const int rem    = N % 16;
  const int gN   = (N + T - 1) / T;
  const int gE   = (E + T - 1) / T;
  const int gNC  = (N * HIDC + T - 1) / T;
  const int gTil = (nTiles + WPB - 1) / WPB;

  // ---- symmetric normalization ----
  k_deg_init <<<gN, T, 0, stream>>>(dinv, N);
  k_deg_accum<<<gE, T, 0, stream>>>(col, ew, dinv, E);
  k_deg_rsqrt<<<gN, T, 0, stream>>>(dinv, N);

  // ---- layer 1: h1 = x @ W1 ; agg1 = selfloop+bias ; scatter ; relu ----
  k_gemm_wmma<<<gTil, T, 0, stream>>>(x, W1, bufA, nTiles, K1);
  if (rem) k_gemm_tail<<<(rem * HIDC + T - 1) / T, T, 0, stream>>>(x, W1, bufA, nTiles * 16, N, K1);
  k_self_bias<<<gNC, T, 0, stream>>>(bufA, dinv, b1, bufB, N);
  k_scatter  <<<gE,  T, 0, stream>>>(row, col, ew, dinv, bufA, bufB, E);
  k_relu     <<<gNC, T, 0, stream>>>(bufB, N * HIDC);

  // ---- layer 2: h2 = relu(agg1) @ W2 ; agg2 = selfloop+bias ; scatter ----
  k_gemm_wmma<<<gTil, T, 0, stream>>>(bufB, W2, bufA, nTiles, HIDC);
  if (rem) k_gemm_tail<<<(rem * HIDC + T - 1) / T, T, 0, stream>>>(bufB, W2, bufA, nTiles * 16, N, HIDC);
  k_self_bias<<<gNC, T, 0, stream>>>(bufA, dinv, b2, bufB, N);
  k_scatter  <<<gE,  T, 0, stream>>>(row, col, ew, dinv, bufA, bufB, E);

  // ---- global mean pool + linear head ----
  k_zero<<<(G * (HIDC + 1) + T - 1) / T, T, 0, stream>>>(pooled, G * (HIDC + 1)); // pooled+cnt contiguous
  k_pool<<<gNC, T, 0, stream>>>(bufB, batch, pooled, cnt, N);
  k_head<<<(G * OUTC + T - 1) / T, T, 0, stream>>>(pooled, cnt, fcW, fcb, (float*)d_out, G, OUTC);
}